// GraphSAGE_76965813944576
// MI455X (gfx1250) — compile-verified
//
#include <hip/hip_runtime.h>

typedef float v2f __attribute__((ext_vector_type(2)));
typedef float v8f __attribute__((ext_vector_type(8)));

#define NFEAT 64
#define WSTRIDE 68          // 64 + 4 pad -> conflict-free ds_load_b64 pattern
#define SCAN_B 256

// ---------------- utility kernels ----------------

__global__ void zero_i_kernel(int* __restrict__ p, int n) {
    int i = blockIdx.x * blockDim.x + threadIdx.x;
    if (i < n) p[i] = 0;
}

// ---------------- CSR build ----------------

__global__ void count_kernel(const int* __restrict__ dst, int* __restrict__ deg, int n_edges) {
    int i = blockIdx.x * blockDim.x + threadIdx.x;
    if (i < n_edges) atomicAdd(&deg[dst[i]], 1);
}

// exclusive scan, stage 1: per-block scan of 256 ints
__global__ void scan_block_kernel(const int* __restrict__ in, int* __restrict__ out,
                                  int* __restrict__ blksum, int n) {
    __shared__ int sh[SCAN_B];
    int t = threadIdx.x;
    int i = blockIdx.x * SCAN_B + t;
    int v = (i < n) ? in[i] : 0;
    sh[t] = v;
    __syncthreads();
    for (int off = 1; off < SCAN_B; off <<= 1) {
        int x = (t >= off) ? sh[t - off] : 0;
        __syncthreads();
        sh[t] += x;
        __syncthreads();
    }
    if (i < n) out[i] = sh[t] - v;                 // exclusive
    if (t == SCAN_B - 1) blksum[blockIdx.x] = sh[t];
}

// stage 2: single-block exclusive scan of block sums (nb <= 1024)
__global__ void scan_sums_kernel(int* __restrict__ blksum, int nb) {
    __shared__ int sh[1024];
    int t = threadIdx.x;
    int v = (t < nb) ? blksum[t] : 0;
    sh[t] = v;
    __syncthreads();
    for (int off = 1; off < 1024; off <<= 1) {
        int x = (t >= off) ? sh[t - off] : 0;
        __syncthreads();
        sh[t] += x;
        __syncthreads();
    }
    if (t < nb) blksum[t] = sh[t] - v;             // exclusive
}

// stage 3: add block offsets; cap row_ofs[n] = E; init placement cursor
__global__ void scan_add_kernel(int* __restrict__ out, const int* __restrict__ blksum,
                                int* __restrict__ cursor, int n, int total) {
    int i = blockIdx.x * blockDim.x + threadIdx.x;
    if (i < n) {
        int v = out[i] + blksum[i / SCAN_B];
        out[i] = v;
        cursor[i] = v;
    }
    if (i == 0) out[n] = total;
}

__global__ void place_kernel(const int* __restrict__ src, const int* __restrict__ dst,
                             int* __restrict__ cursor, int* __restrict__ csr_src, int n_edges) {
    int i = blockIdx.x * blockDim.x + threadIdx.x;
    if (i < n_edges) {
        int d = dst[i];
        int pos = atomicAdd(&cursor[d], 1);
        csr_src[pos] = src[i];
    }
}

__global__ void invdeg_kernel(const int* __restrict__ deg, float* __restrict__ invd, int n) {
    int i = blockIdx.x * blockDim.x + threadIdx.x;
    if (i < n) invd[i] = 1.0f / fmaxf((float)deg[i], 1.0f);
}

// ---------------- gather aggregation (no atomics) ----------------
// One wave per node; each lane owns 2 consecutive feats. Per in-edge the wave
// reads one contiguous 256B row of h (L2-resident). 2x unrolled with split
// accumulators to keep more loads in flight. inv_deg fused into the store.
__global__ void gather_kernel(const float* __restrict__ h,
                              const int* __restrict__ csr_src,
                              const int* __restrict__ row_ofs,
                              const float* __restrict__ invd,
                              float* __restrict__ agg, int n_nodes) {
    int wave = (blockIdx.x * blockDim.x + threadIdx.x) >> 5;
    int lane = threadIdx.x & 31;
    if (wave >= n_nodes) return;
    int beg = row_ofs[wave];
    int end = row_ofs[wave + 1];
    const int f = lane * 2;
    float ax = 0.0f, ay = 0.0f, bx = 0.0f, by = 0.0f;
    int e = beg;
    for (; e + 2 <= end; e += 2) {
        int s0 = csr_src[e];
        int s1 = csr_src[e + 1];
        const v2f v0 = *(const v2f*)(h + (long)s0 * NFEAT + f);
        const v2f v1 = *(const v2f*)(h + (long)s1 * NFEAT + f);
        ax += v0[0]; ay += v0[1];
        bx += v1[0]; by += v1[1];
    }
    if (e < end) {
        int s = csr_src[e];
        const v2f v = *(const v2f*)(h + (long)s * NFEAT + f);
        ax += v[0]; ay += v[1];
    }
    float sc = invd[wave];
    v2f o;
    o[0] = (ax + bx) * sc;
    o[1] = (ay + by) * sc;
    *(v2f*)(agg + (long)wave * NFEAT + f) = o;
}

// ---------------- fused dual-GEMM + bias (+ReLU) via V_WMMA_F32_16X16X4_F32 -----
// out[16-row strip] = relu( h@Wself + agg@Wneigh + bias )   (agg pre-scaled)
// Weights staged TRANSPOSED in LDS (wT[n*68 + k]) so each B fragment is one
// aligned, bank-conflict-free ds_load_b64 directly into an even VGPR pair.
__global__ void sage_gemm_kernel(const float* __restrict__ h,     // [N,64]
                                 const float* __restrict__ agg,   // [N,64] mean-agg
                                 const float* __restrict__ Wself, // [64,64] (k,n)
                                 const float* __restrict__ Wneigh,// [64,64] (k,n)
                                 const float* __restrict__ bias,  // [64]
                                 float* __restrict__ out,         // [N,64]
                                 int n_nodes, int do_relu) {
    __shared__ float wS[NFEAT * WSTRIDE];
    __shared__ float wN[NFEAT * WSTRIDE];
    const int t = threadIdx.x;
    for (int i = t; i < NFEAT * NFEAT; i += blockDim.x) {
        int k = i >> 6;
        int n = i & 63;
        wS[n * WSTRIDE + k] = Wself[i];      // coalesced global read, transposed store
        wN[n * WSTRIDE + k] = Wneigh[i];
    }
    __syncthreads();

    int wave = (blockIdx.x * blockDim.x + t) >> 5;
    int lane = t & 31;
    int row0 = wave * 16;
    if (row0 >= n_nodes) return;            // wave-uniform: EXEC all-1s for WMMA

    const int mrow  = lane & 15;            // A row / B,C,D column lane
    const int khalf = lane >> 4;            // which K-pair this lane holds
    const long row  = row0 + mrow;
    const int kbase = 2 * khalf;            // fragment k offset within K=4 slice

    v2f aS[16], aN[16];
    const float* hrow = h   + row * NFEAT;
    const float* grow = agg + row * NFEAT;
#pragma unroll
    for (int k = 0; k < 16; ++k) {
        int c = 4 * k + kbase;
        aS[k] = *(const v2f*)(hrow + c);
        aN[k] = *(const v2f*)(grow + c);
    }

#pragma unroll
    for (int nt = 0; nt < 4; ++nt) {
        const int col = nt * 16 + mrow;
        const float* bSrow = wS + col * WSTRIDE + kbase;  // per-lane base
        const float* bNrow = wN + col * WSTRIDE + kbase;
        v8f acc = {};
#pragma unroll
        for (int k = 0; k < 16; ++k) {
            const v2f bS = *(const v2f*)(bSrow + 4 * k);  // ds_load_b64, imm offset
            acc = __builtin_amdgcn_wmma_f32_16x16x4_f32(
                false, aS[k], false, bS, (short)0, acc, false, false);
            const v2f bN = *(const v2f*)(bNrow + 4 * k);
            acc = __builtin_amdgcn_wmma_f32_16x16x4_f32(
                false, aN[k], false, bN, (short)0, acc, false, false);
        }
        const float bv = bias[col];
#pragma unroll
        for (int r = 0; r < 8; ++r) {       // C/D layout: row = r + 8*khalf
            float v = acc[r] + bv;
            if (do_relu) v = fmaxf(v, 0.0f);
            out[(long)(row0 + r + 8 * khalf) * NFEAT + col] = v;
        }
    }
}

// ---------------- host orchestration ----------------

static inline int cdiv_i(long a, long b) { return (int)((a + b - 1) / b); }

extern "C" void kernel_launch(void* const* d_in, const int* in_sizes, int n_in,
                              void* d_out, int out_size, void* d_ws, size_t ws_size,
                              hipStream_t stream) {
    const float* x   = (const float*)d_in[0];
    const int*   src = (const int*)d_in[1];
    const int*   dst = (const int*)d_in[2];
    const float* ws0 = (const float*)d_in[3];
    const float* wn0 = (const float*)d_in[4];
    const float* b0  = (const float*)d_in[5];
    const float* wsL = (const float*)d_in[6];   // [4,64,64]
    const float* wnL = (const float*)d_in[7];   // [4,64,64]
    const float* bsL = (const float*)d_in[8];   // [4,64]
    float* out = (float*)d_out;

    const int N = in_sizes[0] / NFEAT;          // 100000 (multiple of 16)
    const int E = in_sizes[1];                  // 1600000
    const int nb = cdiv_i(N, SCAN_B);           // scan blocks (<= 1024)

    // workspace layout (4-byte units, contiguous):
    float* invd   = (float*)d_ws;               // [N]
    float* hA     = invd + N;                   // [N*64]
    float* hB     = hA + (size_t)N * NFEAT;     // [N*64]
    float* agg    = hB + (size_t)N * NFEAT;     // [N*64]
    int*   degi   = (int*)(agg + (size_t)N * NFEAT);  // [N]
    int*   rowofs = degi + N;                   // [N+1]
    int*   cursor = rowofs + (N + 1);           // [N]
    int*   blks   = cursor + N;                 // [1024]
    int*   csr    = blks + 1024;                // [E]

    const int TPB = 256;

    // ---- CSR build (once per call) ----
    zero_i_kernel<<<cdiv_i(N, TPB), TPB, 0, stream>>>(degi, N);
    count_kernel<<<cdiv_i(E, TPB), TPB, 0, stream>>>(dst, degi, E);
    scan_block_kernel<<<nb, SCAN_B, 0, stream>>>(degi, rowofs, blks, N);
    scan_sums_kernel<<<1, 1024, 0, stream>>>(blks, nb);
    scan_add_kernel<<<cdiv_i(N, TPB), TPB, 0, stream>>>(rowofs, blks, cursor, N, E);
    place_kernel<<<cdiv_i(E, TPB), TPB, 0, stream>>>(src, dst, cursor, csr, E);
    invdeg_kernel<<<cdiv_i(N, TPB), TPB, 0, stream>>>(degi, invd, N);

    const int gath_bl = cdiv_i((long)N * 32, TPB);
    const int gemm_bl = cdiv_i((long)(N / 16) * 32, TPB);

    // ---- layer 0: x -> hA (ReLU) ----
    gather_kernel<<<gath_bl, TPB, 0, stream>>>(x, csr, rowofs, invd, agg, N);
    sage_gemm_kernel<<<gemm_bl, TPB, 0, stream>>>(x, agg, ws0, wn0, b0, hA, N, 1);

    // ---- layers 1..4 ----
    const float* hin  = hA;
    float*       hout = hB;
    for (int i = 0; i < 4; ++i) {
        const float* Ws = wsL + (size_t)i * NFEAT * NFEAT;
        const float* Wn = wnL + (size_t)i * NFEAT * NFEAT;
        const float* bi = bsL + (size_t)i * NFEAT;
        float* dstbuf = (i == 3) ? out : hout;
        int relu = (i < 3) ? 1 : 0;

        gather_kernel<<<gath_bl, TPB, 0, stream>>>(hin, csr, rowofs, invd, agg, N);
        sage_gemm_kernel<<<gemm_bl, TPB, 0, stream>>>(hin, agg, Ws, Wn, bi,
                                                      dstbuf, N, relu);
        const float* tswap = hin;
        hin  = hout;
        hout = (float*)tswap;
    }
}